// MIMOGeneralDotProductAttention_25615184953971
// MI455X (gfx1250) — compile-verified
//
#include <hip/hip_runtime.h>
#include <math.h>

typedef __attribute__((ext_vector_type(2))) float v2f;
typedef __attribute__((ext_vector_type(4))) float v4f;
typedef __attribute__((ext_vector_type(8))) float v8f;

#define B_   8
#define A_   6
#define D_   256
#define CHW_ (256 * 32 * 32)      // 262144 elements per (b,k) slice of v
#define ROWS_ (B_ * A_)           // 48 rows of qu / query

// ---------------------------------------------------------------------------
// Kernel 1: query = qu @ W^T + b   via V_WMMA_F32_16X16X4_F32
//   qu: [48, 256] row-major, W: [256, 256] row-major, query: [48, 256]
//   Grid: 48 single-wave blocks; block = (rowTile 0..2, colTile 0..15).
//   A tile = 16x4 of qu, B tile = 4x16 with B[k,n] = W[n,k].
//   Fragment layout (ISA 7.12.2): component i, lane-half h -> K = i + 2h.
// ---------------------------------------------------------------------------
__global__ __launch_bounds__(32)
void proj_wmma_kernel(const float* __restrict__ qu,
                      const float* __restrict__ W,
                      const float* __restrict__ bias,
                      float* __restrict__ query) {
  const int rt   = blockIdx.x >> 4;    // row tile 0..2
  const int ct   = blockIdx.x & 15;    // col tile 0..15
  const int lane = threadIdx.x;
  const int half = lane >> 4;          // lane half (0: lanes 0-15, 1: 16-31)
  const int l    = lane & 15;

  const int arow = rt * 16 + l;        // qu row for A fragment
  const int bcol = ct * 16 + l;        // output col for B fragment (= W row)

  v8f acc = {};
  #pragma unroll 4
  for (int k0 = 0; k0 < D_; k0 += 4) {
    const int kb = k0 + half * 2;
    v2f a, bm;
    a.x  = qu[arow * D_ + kb];
    a.y  = qu[arow * D_ + kb + 1];
    bm.x = W[bcol * D_ + kb];          // B[kb,   bcol] = W[bcol, kb]
    bm.y = W[bcol * D_ + kb + 1];      // B[kb+1, bcol]
    acc = __builtin_amdgcn_wmma_f32_16x16x4_f32(
        /*neg_a=*/false, a, /*neg_b=*/false, bm,
        /*c_mod=*/(short)0, acc, /*reuse_a=*/false, /*reuse_b=*/false);
  }

  // D layout: VGPR i -> M = i + 8*half, N = l   (ISA 7.12.2, 32-bit C/D 16x16)
  const float bv    = bias[bcol];
  const int   rbase = rt * 16 + half * 8;
  #pragma unroll
  for (int i = 0; i < 8; ++i) {
    query[(rbase + i) * D_ + bcol] = acc[i] + bv;
  }
}

// ---------------------------------------------------------------------------
// Kernel 2: attn[b,k,q] = <k[b,k,:], query[b,q,:]>, softmax over k (axis=1).
//   One block per batch. 36 dot products, then 6 softmax columns.
//   Writes attn_sm to workspace (for kernel 3) AND to the tuple output tail.
// ---------------------------------------------------------------------------
__global__ __launch_bounds__(64)
void attn_softmax_kernel(const float* __restrict__ kmat,
                         const float* __restrict__ query,
                         float* __restrict__ attn_ws,
                         float* __restrict__ attn_out) {
  __shared__ float s[A_ * A_];
  const int b   = blockIdx.x;
  const int tid = threadIdx.x;

  if (tid < A_ * A_) {
    const int kk = tid / A_;
    const int q  = tid % A_;
    const float* kr = kmat  + ((size_t)b * A_ + kk) * D_;
    const float* qr = query + ((size_t)b * A_ + q)  * D_;
    float dot = 0.f;
    for (int d = 0; d < D_; ++d) dot = fmaf(kr[d], qr[d], dot);
    s[tid] = dot;
  }
  __syncthreads();

  if (tid < A_) {
    const int q = tid;
    float m = s[q];
    #pragma unroll
    for (int kk = 1; kk < A_; ++kk) m = fmaxf(m, s[kk * A_ + q]);
    float e[A_];
    float sum = 0.f;
    #pragma unroll
    for (int kk = 0; kk < A_; ++kk) {
      e[kk] = __expf(s[kk * A_ + q] - m);
      sum += e[kk];
    }
    const float inv = 1.f / sum;
    #pragma unroll
    for (int kk = 0; kk < A_; ++kk) {
      const float r = e[kk] * inv;
      attn_ws [b * A_ * A_ + kk * A_ + q] = r;
      attn_out[b * A_ * A_ + kk * A_ + q] = r;
    }
  }
}

// ---------------------------------------------------------------------------
// Kernel 3 (the bandwidth-bound 99%): out[b,q,chw] = sum_k attn[b,k,q]*v[b,k,chw]
//   Each thread owns one float4 position: 6x B128 NT loads, 36 FMAs-on-vec4,
//   6x B128 NT stores. 100 MB total traffic -> ~4.3 us floor at 23.3 TB/s.
//   Attention weights are per-batch uniform -> scalar (SGPR) loads.
// ---------------------------------------------------------------------------
__global__ __launch_bounds__(256)
void weighted_sum_kernel(const float* __restrict__ v,
                         const float* __restrict__ attn_ws,
                         float* __restrict__ out) {
  const int b    = blockIdx.x >> 8;            // 8 batches * 256 blocks each
  const int blk  = blockIdx.x & 255;
  const int pos4 = blk * 256 + (int)threadIdx.x;   // float4 index in [0,65536)
  const size_t elem = (size_t)pos4 * 4;

  // 6x6 attention matrix for this batch; uniform address -> SGPRs
  float w[A_ * A_];
  #pragma unroll
  for (int i = 0; i < A_ * A_; ++i) w[i] = attn_ws[b * A_ * A_ + i];

  const v4f zero = {0.f, 0.f, 0.f, 0.f};
  v4f acc[A_];
  #pragma unroll
  for (int q = 0; q < A_; ++q) acc[q] = zero;

  #pragma unroll
  for (int kk = 0; kk < A_; ++kk) {
    const v4f* vp =
        (const v4f*)(v + ((size_t)(b * A_ + kk) * CHW_ + elem));
    const v4f vv = __builtin_nontemporal_load(vp);
    #pragma unroll
    for (int q = 0; q < A_; ++q) acc[q] += w[kk * A_ + q] * vv;
  }

  #pragma unroll
  for (int q = 0; q < A_; ++q) {
    v4f* op = (v4f*)(out + ((size_t)(b * A_ + q) * CHW_ + elem));
    __builtin_nontemporal_store(acc[q], op);
  }
}

// ---------------------------------------------------------------------------
extern "C" void kernel_launch(void* const* d_in, const int* in_sizes, int n_in,
                              void* d_out, int out_size, void* d_ws, size_t ws_size,
                              hipStream_t stream) {
  const float* qu   = (const float*)d_in[0];   // [8,6,256]
  const float* kmat = (const float*)d_in[1];   // [8,6,256]
  const float* v    = (const float*)d_in[2];   // [8,6,256,32,32]
  const float* W    = (const float*)d_in[3];   // [256,256]
  const float* bias = (const float*)d_in[4];   // [256]

  float* out      = (float*)d_out;                       // [8,6,256,32,32]
  float* attn_out = out + (size_t)B_ * A_ * CHW_;        // [8,6,6] tuple tail

  float* query_ws = (float*)d_ws;                        // 48*256 floats
  float* attn_ws  = query_ws + ROWS_ * D_;               // 288 floats

  // 1) query = qu @ W^T + b  (WMMA f32 16x16x4; 3x16 tiles, 1 wave each)
  proj_wmma_kernel<<<dim3(48), dim3(32), 0, stream>>>(qu, W, bias, query_ws);

  // 2) scores + softmax over k  (tiny; one block per batch)
  attn_softmax_kernel<<<dim3(B_), dim3(64), 0, stream>>>(kmat, query_ws,
                                                         attn_ws, attn_out);

  // 3) streaming weighted sum (the bandwidth-bound bulk)
  weighted_sum_kernel<<<dim3(B_ * 256), dim3(256), 0, stream>>>(v, attn_ws, out);
}